// StructuredAttention_38036230373747
// MI455X (gfx1250) — compile-verified
//
#include <hip/hip_runtime.h>
#include <hip/hip_bf16.h>
#include <math.h>

// ---------------------------------------------------------------------------
// StructuredAttention (matrix-tree attention) for MI455X / gfx1250.
// All dense GEMMs use V_WMMA_F32_16X16X4_F32 (full-fp32 matrix pipe) because
// the pipeline feeds exp() + a dense matrix inversion (precision critical).
// GEMM: 32x64 register macro-tile per wave (8 independent WMMAs / k-step),
// operand vectorization (b64 loads) chosen at compile time per call site,
// all operand addressing via pointer-bumping (no in-loop address math).
// Inversion: in-place Gauss-Jordan, one 1024-thread WG per batch, pivot
// row/col staged in LDS, float4 rank-1 updates (L2-resident, 4MB/matrix).
// ---------------------------------------------------------------------------

typedef __attribute__((ext_vector_type(2))) float v2f;
typedef __attribute__((ext_vector_type(8))) float v8f;

#define B_    16
#define K_    1024
#define D_    768
#define SEM_  384
#define STR_  384
#define CAT_  1152

// Workspace offsets (floats). High-water = 75,546,624 floats (~302 MB).
static const long long F_E    = 0LL;          // e_i           [B*K,384]
static const long long F_D    = 6291456LL;    // d_i (later T) [B*K,384]
static const long long F_UP   = 12582912LL;   // u_j           [B*K,384]
static const long long F_UC   = 18874368LL;   // u_k           [B*K,384]
static const long long F_A    = 25165824LL;   // A scores      [B,K,K]
static const long long F_L    = 41943040LL;   // L_bar -> Linv [B,K,K]
static const long long F_CS   = 58720256LL;   // col sums      [B,K]
static const long long F_FI   = 58736640LL;   // f_i           [B,K]
static const long long F_DIAG = 58753024LL;   // diag(Linv)    [B,K]
static const long long F_AIK  = 58769408LL;   // a_ik          [B,K,K]
static const long long F_SSR  = F_UP;         // ssr [B,1025,384] (Up/Uc dead)
static const long long F_CAT  = F_A;          // concat [B*K,1152] (A/L dead)

// ---------------------------------------------------------------------------
// 0) channel split:  e_i / d_i gather from x
// ---------------------------------------------------------------------------
__global__ void k_gather_ed(const float* __restrict__ x,
                            float* __restrict__ e, float* __restrict__ d,
                            int total) {
  int i = blockIdx.x * blockDim.x + threadIdx.x;
  if (i >= total) return;
  int c = i % SEM_;
  long long row = i / SEM_;
  const float* xr = x + row * (long long)D_;
  int ce = (c < 192) ? c       : c + 192;   // e: [0:192] + [384:576]
  int cd = (c < 192) ? c + 192 : c + 384;   // d: [192:384] + [576:768]
  e[i] = xr[ce];
  d[i] = xr[cd];
}

// ---------------------------------------------------------------------------
// Generic batched strided GEMM on the f32 WMMA pipe.
//   C[bat][m, cColOff+n] = epi( sum_k A[bat][m*aRS + k*aCS] *
//                                     B[bat][k*bRS + n*bCS] + bias[n] )
// One wave computes a 32x64 macro-tile (2 M-tiles x 4 N-tiles -> 8 WMMA
// accumulators).  Wave-uniform early exit keeps EXEC all-ones (WMMA req).
// VECA/VECB are compile-time: K-contiguous operands with even leading
// stride use one b64 load per fragment (fragment K offset kk+2*hi is even).
// All operand addresses are pre-computed per-lane pointers bumped by a
// constant stride each k-step -> no address math or branches in the loop.
// Scalar K tail handles Kc % 4 != 0 (the a_ki @ ssr GEMM has Kc = 1025).
// f32 WMMA fragment layout (ISA 7.12.2):
//   A (16x4):  lane<16 -> {A[r,k0],A[r,k0+1]}, lane>=16 -> k0+2 offset
//   B (4x16):  mirrored K split;  C/D: row = v + 8*(lane>>4), col = lane&15
// ---------------------------------------------------------------------------
enum { EPI_NONE = 0, EPI_TANH = 1, EPI_EXPD = 2 };

template <int EPI, bool VECA, bool VECB>
__global__ void k_gemm_wmma(const float* __restrict__ A, long long aBat, int aRS, int aCS,
                            const float* __restrict__ Bm, long long bBat, int bRS, int bCS,
                            float* __restrict__ C, long long cBat, int cRS, int cColOff,
                            const float* __restrict__ bias,
                            int M, int N, int Kc, int tilesM2, int tilesN4, int totalWaves) {
  int wid = (int)((blockIdx.x * blockDim.x + threadIdx.x) >> 5);
  if (wid >= totalWaves) return;                  // wave-uniform
  int lane = threadIdx.x & 31;
  int hi   = lane >> 4;
  int l15  = lane & 15;

  int tilesPB = tilesM2 * tilesN4;
  int bat = wid / tilesPB;
  int rem = wid - bat * tilesPB;
  int tm  = rem / tilesN4;                        // 32-row macro tile
  int tn  = rem - tm * tilesN4;                   // 64-col macro tile

  const float* Ab = A  + (long long)bat * aBat;
  const float* Bb = Bm + (long long)bat * bBat;
  float*       Cb = C  + (long long)bat * cBat;

  int r0 = tm * 32 + l15;                         // A rows for the 2 M-tiles
  int c0 = tn * 64 + l15;                         // B col for N-tile j: c0+16j

  // Per-lane operand pointers (include the lane's K offset 2*hi); bumped by
  // a loop-invariant stride each k-step.
  const float* pa0;
  const float* pa1;
  long long aStep;
  if (VECA) {
    pa0 = Ab + (long long)r0 * aRS + 2 * hi;
    pa1 = Ab + (long long)(r0 + 16) * aRS + 2 * hi;
    aStep = 4;
  } else {
    pa0 = Ab + (long long)r0 * aRS + (long long)(2 * hi) * aCS;
    pa1 = Ab + (long long)(r0 + 16) * aRS + (long long)(2 * hi) * aCS;
    aStep = 4LL * aCS;
  }
  const float* pb[4];
  long long bStep;
#pragma unroll
  for (int j = 0; j < 4; ++j) {
    long long cb = (long long)(c0 + 16 * j) * bCS;
    pb[j] = VECB ? (Bb + cb + 2 * hi)
                 : (Bb + cb + (long long)(2 * hi) * bRS);
  }
  bStep = VECB ? 4LL : 4LL * bRS;

  v8f acc[2][4];
#pragma unroll
  for (int i = 0; i < 2; ++i)
#pragma unroll
    for (int j = 0; j < 4; ++j)
      acc[i][j] = (v8f){0.f, 0.f, 0.f, 0.f, 0.f, 0.f, 0.f, 0.f};

  int K4 = Kc & ~3;
#pragma unroll 2
  for (int kk = 0; kk < K4; kk += 4) {
    v2f af[2], bf[4];
    if (VECA) {
      af[0] = *(const v2f*)pa0;
      af[1] = *(const v2f*)pa1;
    } else {
      af[0].x = pa0[0];
      af[0].y = pa0[aCS];
      af[1].x = pa1[0];
      af[1].y = pa1[aCS];
    }
#pragma unroll
    for (int j = 0; j < 4; ++j) {
      if (VECB) {
        bf[j] = *(const v2f*)pb[j];
      } else {
        bf[j].x = pb[j][0];
        bf[j].y = pb[j][bRS];
      }
    }
    pa0 += aStep; pa1 += aStep;
#pragma unroll
    for (int j = 0; j < 4; ++j) pb[j] += bStep;

#pragma unroll
    for (int i = 0; i < 2; ++i)
#pragma unroll
      for (int j = 0; j < 4; ++j)
        acc[i][j] = __builtin_amdgcn_wmma_f32_16x16x4_f32(
            false, (i == 0) ? af[0] : af[1], false, bf[j],
            (short)0, acc[i][j], false, false);
  }

  for (int kk = K4; kk < Kc; ++kk) {              // scalar K tail (Kc = 1025)
    float bv[4];
#pragma unroll
    for (int j = 0; j < 4; ++j)
      bv[j] = Bb[(long long)kk * bRS + (long long)(c0 + 16 * j) * bCS];
#pragma unroll
    for (int i = 0; i < 2; ++i) {
#pragma unroll
      for (int v = 0; v < 8; ++v) {
        int r = tm * 32 + i * 16 + v + 8 * hi;
        float av = Ab[(long long)r * aRS + (long long)kk * aCS];
#pragma unroll
        for (int j = 0; j < 4; ++j) acc[i][j][v] += av * bv[j];
      }
    }
  }

#pragma unroll
  for (int j = 0; j < 4; ++j) {
    int col = c0 + 16 * j;
    float bb = bias ? bias[col] : 0.f;
#pragma unroll
    for (int i = 0; i < 2; ++i) {
#pragma unroll
      for (int v = 0; v < 8; ++v) {
        int row = tm * 32 + i * 16 + v + 8 * hi;
        float xv = acc[i][j][v] + bb;
        if (EPI == EPI_TANH) xv = tanhf(xv);
        if (EPI == EPI_EXPD) xv = (row == col) ? 0.0f : expf(xv);  // exp(f)*(1-I)
        Cb[(long long)row * cRS + cColOff + col] = xv;
      }
    }
  }
}

// ---------------------------------------------------------------------------
// f_i = exp(d_i @ fi_w)
// ---------------------------------------------------------------------------
__global__ void k_fi(const float* __restrict__ d, const float* __restrict__ fiw,
                     float* __restrict__ fi, int rows) {
  int r = blockIdx.x * blockDim.x + threadIdx.x;
  if (r >= rows) return;
  const float* dr = d + (long long)r * STR_;
  float s = 0.f;
  for (int i = 0; i < STR_; ++i) s = fmaf(dr[i], fiw[i], s);
  fi[r] = expf(s);
}

// col_sum[b,k] = sum_j A[b,j,k]   (coalesced over k)
__global__ void k_colsum(const float* __restrict__ A, float* __restrict__ cs) {
  int bpb = K_ / 256;
  int b = blockIdx.x / bpb;
  int k = (blockIdx.x % bpb) * 256 + threadIdx.x;
  const float* Ab = A + (long long)b * K_ * K_;
  float s = 0.f;
  for (int j = 0; j < K_; ++j) s += Ab[(long long)j * K_ + k];
  cs[b * K_ + k] = s;
}

// L_bar: row0 = f_i;  diag = col_sum;  offdiag = -A
__global__ void k_buildL(const float* __restrict__ A, const float* __restrict__ cs,
                         const float* __restrict__ fi, float* __restrict__ L,
                         long long total) {
  long long idx = (long long)blockIdx.x * blockDim.x + threadIdx.x;
  if (idx >= total) return;
  int j = (int)(idx % K_);
  long long t = idx / K_;
  int i = (int)(t % K_);
  int b = (int)(t / K_);
  float v;
  if (i == 0)       v = fi[b * K_ + j];
  else if (i == j)  v = cs[b * K_ + j];
  else              v = -A[idx];
  L[idx] = v;
}

// ---------------------------------------------------------------------------
// In-place Gauss-Jordan inversion of one 1024x1024 matrix per workgroup.
// Pivot column + scaled pivot row staged in LDS; rowp[p]=1+piv makes the
// rank-1 update branch-free including the pivot column (M[i,p] -> -f*piv).
// ---------------------------------------------------------------------------
__global__ void __launch_bounds__(1024) k_invert(float* __restrict__ Lg) {
  __shared__ float colp[K_];
  __shared__ float rowp[K_];
  float* M = Lg + (long long)blockIdx.x * K_ * K_;
  int t = threadIdx.x;
  for (int p = 0; p < K_; ++p) {
    colp[t] = M[(long long)t * K_ + p];
    __syncthreads();
    float piv = 1.0f / colp[p];
    float rv = M[(long long)p * K_ + t] * piv;        // scale pivot row
    rowp[t] = (t == p) ? (1.0f + piv) : rv;
    M[(long long)p * K_ + t] = (t == p) ? piv : rv;
    __syncthreads();
    if (t != p) {
      float f = colp[t];
      float* Mr = M + (long long)t * K_;
      for (int j = 0; j < K_; j += 4) {
        float4 mv = *(float4*)(Mr + j);
        mv.x = fmaf(-f, rowp[j + 0], mv.x);
        mv.y = fmaf(-f, rowp[j + 1], mv.y);
        mv.z = fmaf(-f, rowp[j + 2], mv.z);
        mv.w = fmaf(-f, rowp[j + 3], mv.w);
        *(float4*)(Mr + j) = mv;
      }
    }
    __syncthreads();
  }
}

// d0 = f_i * Linv[:,:,0] (written as a_ki column 0);  diag = Linv[k,k]
__global__ void k_d0diag(const float* __restrict__ Linv, const float* __restrict__ fi,
                         float* __restrict__ diag, float* __restrict__ aki, int total) {
  int i = blockIdx.x * blockDim.x + threadIdx.x;
  if (i >= total) return;                      // i = b*K + k
  int b = i / K_, k = i % K_;
  const float* Lb = Linv + (long long)b * K_ * K_;
  float d0 = fi[i] * Lb[(long long)k * K_];
  diag[i] = Lb[(long long)k * K_ + k];
  aki[(long long)b * K_ * (K_ + 1) + (long long)k * (K_ + 1)] = d0;
}

// a_ik[b,j,k] = A * ((k!=0 ? Linv[k,k] : 0) - (j!=0 ? Linv[k,j] : 0))
// also scatters a_ki[b,k,j+1] = a_ik[b,j,k] straight into d_out.
__global__ void k_marg(const float* __restrict__ A, const float* __restrict__ Linv,
                       const float* __restrict__ diag,
                       float* __restrict__ aik, float* __restrict__ aki,
                       long long total) {
  long long idx = (long long)blockIdx.x * blockDim.x + threadIdx.x;
  if (idx >= total) return;
  int k = (int)(idx % K_);
  long long t = idx / K_;
  int j = (int)(t % K_);
  int b = (int)(t / K_);
  const float* Lb = Linv + (long long)b * K_ * K_;
  float a    = A[idx];
  float term = (k != 0) ? diag[b * K_ + k] : 0.f;
  float sub  = (j != 0) ? Lb[(long long)k * K_ + j] : 0.f;
  float v = a * (term - sub);
  aik[idx] = v;
  aki[(long long)b * K_ * (K_ + 1) + (long long)k * (K_ + 1) + j + 1] = v;
}

// ssr[b,0,:] = exparam ; ssr[b,1+t,:] = e[b,t,:]
__global__ void k_ssr(const float* __restrict__ e, const float* __restrict__ exparam,
                      float* __restrict__ ssr, int total) {
  int i = blockIdx.x * blockDim.x + threadIdx.x;
  if (i >= total) return;
  int c = i % SEM_;
  int t = i / SEM_;
  int j = t % (K_ + 1);
  int b = t / (K_ + 1);
  ssr[i] = (j == 0) ? exparam[c]
                    : e[((long long)b * K_ + (j - 1)) * SEM_ + c];
}

// cat[:, 0:384] = e   (si/ci GEMMs fill cols 384:768 / 768:1152 directly)
__global__ void k_cate(const float* __restrict__ e, float* __restrict__ cat, int total) {
  int i = blockIdx.x * blockDim.x + threadIdx.x;
  if (i >= total) return;
  int c = i % SEM_;
  int r = i / SEM_;
  cat[(long long)r * CAT_ + c] = e[i];
}

// ---------------------------------------------------------------------------
extern "C" void kernel_launch(void* const* d_in, const int* in_sizes, int n_in,
                              void* d_out, int out_size, void* d_ws, size_t ws_size,
                              hipStream_t stream) {
  (void)in_sizes; (void)n_in; (void)out_size; (void)ws_size;
  const float* x     = (const float*)d_in[0];
  const float* Wp_w  = (const float*)d_in[1];
  const float* Wp_b  = (const float*)d_in[2];
  const float* Wc_w  = (const float*)d_in[3];
  const float* Wc_b  = (const float*)d_in[4];
  const float* fi_w  = (const float*)d_in[5];
  const float* Wa_w  = (const float*)d_in[6];
  const float* expar = (const float*)d_in[7];
  const float* Wr_w  = (const float*)d_in[8];
  const float* Wr_b  = (const float*)d_in[9];

  float* ws   = (float*)d_ws;
  float* e    = ws + F_E;
  float* dmat = ws + F_D;
  float* Up   = ws + F_UP;
  float* Uc   = ws + F_UC;
  float* Aw   = ws + F_A;
  float* Lw   = ws + F_L;
  float* cs   = ws + F_CS;
  float* fi   = ws + F_FI;
  float* diag = ws + F_DIAG;
  float* aik  = ws + F_AIK;
  float* Tm   = ws + F_D;    // reuse: d_i dead after f_i/U GEMMs
  float* ssr  = ws + F_SSR;  // reuse: Up/Uc dead after score GEMM
  float* cat  = ws + F_CAT;  // reuse: A/L dead after marginals

  float* r_out = (float*)d_out;                                  // [B,K,SEM]
  float* aki   = (float*)d_out + (long long)B_ * K_ * SEM_;      // [B,K,K+1]

  const int BK = B_ * K_;

#define GEMM(EPI, VA, VB, Ap, aB, aRS, aCS, Bp, bB, bRS, bCS, Cp, cB, cRS, cCol, bias, M, N, Kc, nb) \
  do {                                                                                          \
    int tM = (M) / 32, tN = (N) / 64;                                                           \
    int waves = (nb) * tM * tN;                                                                 \
    int blocks = (waves + 7) / 8;                                                               \
    k_gemm_wmma<EPI, VA, VB><<<blocks, 256, 0, stream>>>(Ap, aB, aRS, aCS, Bp, bB, bRS, bCS,    \
                                                         Cp, cB, cRS, cCol, bias, M, N, Kc,     \
                                                         tM, tN, waves);                        \
  } while (0)

  // 1) gather e_i / d_i
  { int total = BK * SEM_;
    k_gather_ed<<<(total + 255) / 256, 256, 0, stream>>>(x, e, dmat, total); }

  // 2) u_j = tanh(d @ Wp + b),  u_k = tanh(d @ Wc + b)
  GEMM(EPI_TANH, true, false, dmat, 0LL, STR_, 1, Wp_w, 0LL, STR_, 1,
       Up, 0LL, STR_, 0, Wp_b, BK, STR_, STR_, 1);
  GEMM(EPI_TANH, true, false, dmat, 0LL, STR_, 1, Wc_w, 0LL, STR_, 1,
       Uc, 0LL, STR_, 0, Wc_b, BK, STR_, STR_, 1);

  // 3) f_i = exp(d @ fi_w)
  k_fi<<<(BK + 255) / 256, 256, 0, stream>>>(dmat, fi_w, fi, BK);

  // 4) T = u_j @ Wa   (overwrites d_i slot)
  GEMM(EPI_NONE, true, false, Up, 0LL, STR_, 1, Wa_w, 0LL, STR_, 1,
       Tm, 0LL, STR_, 0, (const float*)nullptr, BK, STR_, STR_, 1);

  // 5) A = exp(T @ u_k^T) with zero diagonal (batched GEMM-NT, both b64 paths)
  GEMM(EPI_EXPD, true, true, Tm, (long long)K_ * STR_, STR_, 1,
       Uc, (long long)K_ * STR_, 1, STR_,
       Aw, (long long)K_ * K_, K_, 0, (const float*)nullptr,
       K_, K_, STR_, B_);

  // 6) Laplacian build
  k_colsum<<<B_ * (K_ / 256), 256, 0, stream>>>(Aw, cs);
  { long long total = (long long)B_ * K_ * K_;
    k_buildL<<<(unsigned)((total + 255) / 256), 256, 0, stream>>>(Aw, cs, fi, Lw, total); }

  // 7) batched dense inversion (Gauss-Jordan, in place)
  k_invert<<<B_, 1024, 0, stream>>>(Lw);

  // 8) marginals: d0 (a_ki col 0), diag, a_ik, a_ki
  k_d0diag<<<(BK + 255) / 256, 256, 0, stream>>>(Lw, fi, diag, aki, BK);
  { long long total = (long long)B_ * K_ * K_;
    k_marg<<<(unsigned)((total + 255) / 256), 256, 0, stream>>>(Aw, Lw, diag, aik, aki, total); }

  // 9) ssr = [exparam ; e_i]
  { int total = B_ * (K_ + 1) * SEM_;
    k_ssr<<<(total + 255) / 256, 256, 0, stream>>>(e, expar, ssr, total); }

  // 10) si = a_ki @ ssr  -> cat[:,384:768]   (Kc=1025: odd aRS -> scalar
  //     operand path + scalar K tail; no misaligned b64)
  GEMM(EPI_NONE, false, false, aki, (long long)K_ * (K_ + 1), K_ + 1, 1,
       ssr, (long long)(K_ + 1) * SEM_, SEM_, 1,
       cat, (long long)K_ * CAT_, CAT_, SEM_, (const float*)nullptr,
       K_, SEM_, K_ + 1, B_);

  // 11) ci = a_ik @ e   -> cat[:,768:1152]
  GEMM(EPI_NONE, true, false, aik, (long long)K_ * K_, K_, 1,
       e, (long long)K_ * SEM_, SEM_, 1,
       cat, (long long)K_ * CAT_, CAT_, 2 * SEM_, (const float*)nullptr,
       K_, SEM_, K_, B_);

  // 12) cat[:,0:384] = e
  { int total = BK * SEM_;
    k_cate<<<(total + 255) / 256, 256, 0, stream>>>(e, cat, total); }

  // 13) r_i = tanh(cat @ Wr + b)
  GEMM(EPI_TANH, true, false, cat, 0LL, CAT_, 1, Wr_w, 0LL, SEM_, 1,
       r_out, 0LL, SEM_, 0, Wr_b, BK, SEM_, CAT_, 1);

#undef GEMM
}